// BatasMemristorTorch_57002805952678
// MI455X (gfx1250) — compile-verified
//
#include <hip/hip_runtime.h>
#include <hip/hip_bf16.h>

// out[i] = Vin[i] / resistance, resistance = RON*(w/D) + ROFF*(1 - w/D).
// Pure HBM-streaming kernel: 256 MiB total traffic, floor ~11.5 us @ 23.3 TB/s.
// Strategy: uniform scalar-path parameter loads + two fp32 divides per thread,
// then 4x-unrolled 128-bit nontemporal loads/stores (TH=NT) per thread.

typedef float v4f __attribute__((ext_vector_type(4)));

__global__ __launch_bounds__(256) void BatasMemristor_current_kernel(
    const float* __restrict__ vin,
    const float* __restrict__ pRON,
    const float* __restrict__ pROFF,
    const float* __restrict__ pD,
    const float* __restrict__ pW,
    float* __restrict__ out,
    int n)
{
    // Wave-uniform parameter reads -> s_load_b32; full-precision scalar setup
    // executed once per thread (cost amortized over 16 elements/thread).
    const float wD  = pW[0] / pD[0];
    const float res = pRON[0] * wD + pROFF[0] * (1.0f - wD);
    const float inv = 1.0f / res;

    const int n4 = n >> 2;  // number of float4 chunks
    constexpr int UNROLL = 4;

    const v4f* __restrict__ vin4 = (const v4f*)vin;
    v4f* __restrict__ out4 = (v4f*)out;

    // Coalesced mapping: within each of the UNROLL steps, consecutive threads
    // touch consecutive float4s -> wave32 covers a contiguous 512B span per
    // global_load_b128 clause.
    const int base = (int)(blockIdx.x * blockDim.x) * UNROLL + (int)threadIdx.x;

#pragma unroll
    for (int k = 0; k < UNROLL; ++k) {
        const int idx = base + k * (int)blockDim.x;
        if (idx < n4) {
            // Streaming data, touched exactly once: non-temporal on both sides
            // so the 256 MiB stream does not thrash the 192 MB L2.
            v4f v = __builtin_nontemporal_load(vin4 + idx);
            v4f r;
            r.x = v.x * inv;
            r.y = v.y * inv;
            r.z = v.z * inv;
            r.w = v.w * inv;
            __builtin_nontemporal_store(r, out4 + idx);
        }
    }

    // Scalar tail for n not divisible by 4 (N = 2^25 is, but stay correct).
    if (blockIdx.x == 0 && threadIdx.x == 0) {
        for (int i = n4 << 2; i < n; ++i) {
            out[i] = vin[i] * inv;
        }
    }
}

extern "C" void kernel_launch(void* const* d_in, const int* in_sizes, int n_in,
                              void* d_out, int out_size, void* d_ws, size_t ws_size,
                              hipStream_t stream) {
    const float* vin   = (const float*)d_in[0];
    const float* pRON  = (const float*)d_in[1];
    const float* pROFF = (const float*)d_in[2];
    const float* pD    = (const float*)d_in[3];
    const float* pW    = (const float*)d_in[4];
    float* out = (float*)d_out;

    const int n = in_sizes[0];           // 2^25
    const int n4 = n >> 2;               // float4 chunks
    constexpr int THREADS = 256;         // 8 wave32s per block
    constexpr int UNROLL = 4;
    const int chunksPerBlock = THREADS * UNROLL;             // 1024 float4 / block
    const int blocks = (n4 + chunksPerBlock - 1) / chunksPerBlock;  // 8192 for N=2^25

    BatasMemristor_current_kernel<<<blocks, THREADS, 0, stream>>>(
        vin, pRON, pROFF, pD, pW, out, n);
}